// GNNNet_36515811951272
// MI455X (gfx1250) — compile-verified
//
#include <hip/hip_runtime.h>
#include <hip/hip_bf16.h>

typedef __attribute__((ext_vector_type(2))) float v2f;
typedef __attribute__((ext_vector_type(8))) float v8f;

#define HDIM 128

// ---------------------------------------------------------------------------
// Dense GEMM: C[nrows x 128] = A[nrows x 128] @ W[128 x 128], fp32 WMMA.
// Block = 256 threads = 8 waves; block covers 128 rows (16 rows per wave).
// W is staged in LDS packed so each lane's B fragment (K-pair {kk,kk+1} for
// its N column) is a single contiguous 8B read, bank-conflict free.
//   wpack[(k0*128 + n)*4 + j] = W[(4*k0 + j)*128 + n],  k0 in [0,32), j in [0,4)
// WMMA f32 16x16x4 fragment layout (ISA 7.12.2):
//   A: lane<16 -> row m=lane, K={0,1}; lane>=16 -> row m=lane-16, K={2,3}
//   B: lane<16 -> col n=lane, K={0,1}; lane>=16 -> col n=lane-16, K={2,3}
//   C: VGPR v, lane<16 -> (row v,   col lane); lane>=16 -> (row v+8, col lane-16)
// ---------------------------------------------------------------------------
__global__ __launch_bounds__(256) void gcn_gemm_wmma(const float* __restrict__ A,
                                                     const float* __restrict__ W,
                                                     float* __restrict__ C,
                                                     int nrows) {
    __shared__ float wpack[HDIM * HDIM];

    // Cooperative pack of W into LDS (coalesced global reads).
    for (int idx = threadIdx.x; idx < HDIM * HDIM; idx += 256) {
        int k = idx >> 7;        // 0..127
        int n = idx & 127;
        wpack[((((k >> 2) << 7) + n) << 2) + (k & 3)] = W[idx];
    }
    __syncthreads();

    const int wave = threadIdx.x >> 5;
    const int lane = threadIdx.x & 31;
    const int half = lane >> 4;       // 0 or 1
    const int m    = lane & 15;       // row (A) / col (B) within tile
    const int kk   = half << 1;       // K sub-offset: 0 or 2

    const int r0 = blockIdx.x * 128 + wave * 16;
    if (r0 >= nrows) return;          // wave-uniform: EXEC stays all-ones below

    v8f acc[8] = {};                  // 16x128 strip = 8 tiles of 16x16

    const float* arow = A + (size_t)(r0 + m) * HDIM + kk;

    for (int k0 = 0; k0 < 32; ++k0) {
        // A fragment: 2 consecutive fp32 (8B aligned since kk is even)
        const v2f a = *(const v2f*)(arow + (k0 << 2));
        const float* wb = &wpack[(k0 << 9) + kk];
#pragma unroll
        for (int t = 0; t < 8; ++t) {
            const int n = (t << 4) + m;
            const v2f b = *(const v2f*)(wb + (n << 2));
            acc[t] = __builtin_amdgcn_wmma_f32_16x16x4_f32(
                false, a, false, b, (short)0, acc[t], false, false);
        }
    }

    // Store C strip (coalesced 64B half-wave chunks).
#pragma unroll
    for (int t = 0; t < 8; ++t) {
        const int col = (t << 4) + m;
#pragma unroll
        for (int v = 0; v < 8; ++v) {
            const int row = r0 + v + (half << 3);
            C[(size_t)row * HDIM + col] = acc[t][v];
        }
    }
}

// ---------------------------------------------------------------------------
// Helper kernels
// ---------------------------------------------------------------------------
__global__ void zero_kernel(float* __restrict__ p, size_t n) {
    size_t i = (size_t)blockIdx.x * blockDim.x + threadIdx.x;
    if (i < n) p[i] = 0.0f;
}

__global__ void count_deg_kernel(const int* __restrict__ dst,
                                 float* __restrict__ deg, int E) {
    int e = blockIdx.x * blockDim.x + threadIdx.x;
    if (e < E) atomicAdd(&deg[dst[e]], 1.0f);   // exact: counts << 2^24
}

__global__ void dinv_kernel(float* __restrict__ deg_dinv, int N) {
    int i = blockIdx.x * blockDim.x + threadIdx.x;
    if (i < N) deg_dinv[i] = rsqrtf(deg_dinv[i] + 1.0f);  // +1 = self-loop
}

__global__ void norm_kernel(const int* __restrict__ src, const int* __restrict__ dst,
                            const float* __restrict__ dinv, float* __restrict__ norm,
                            int E) {
    int e = blockIdx.x * blockDim.x + threadIdx.x;
    if (e < E) norm[e] = dinv[src[e]] * dinv[dst[e]];
}

// One wave32 per edge: 512B coalesced gather of h[src], scaled, 128 L2-resident
// fp32 atomic adds into out[dst]. h (51 MB) fits in the 192 MB L2.
__global__ __launch_bounds__(256) void scatter_add_kernel(const float* __restrict__ h,
                                                          const int* __restrict__ src,
                                                          const int* __restrict__ dst,
                                                          const float* __restrict__ norm,
                                                          float* __restrict__ out,
                                                          int E) {
    const int e = blockIdx.x * 8 + (threadIdx.x >> 5);
    if (e >= E) return;
    const int lane = threadIdx.x & 31;
    const int s = src[e];
    const int d = dst[e];
    const float w = norm[e];

    const float4 v = *(const float4*)(h + (size_t)s * HDIM + (lane << 2));
    float* o = out + (size_t)d * HDIM + (lane << 2);
    atomicAdd(o + 0, v.x * w);
    atomicAdd(o + 1, v.y * w);
    atomicAdd(o + 2, v.z * w);
    atomicAdd(o + 3, v.w * w);
}

// acc[i,:] = relu(acc[i,:] + dinv[i]^2 * h2[i,:] + b)   (self-loop + bias + relu)
__global__ void finalize_kernel(float* __restrict__ acc, const float* __restrict__ h2,
                                const float* __restrict__ dinv,
                                const float* __restrict__ b, int N) {
    size_t idx = (size_t)blockIdx.x * blockDim.x + threadIdx.x;
    if (idx >= (size_t)N * HDIM) return;
    const int i = (int)(idx >> 7);
    const int j = (int)(idx & 127);
    const float di = dinv[i];
    const float v = acc[idx] + di * di * h2[idx] + b[j];
    acc[idx] = fmaxf(v, 0.0f);
}

// ---------------------------------------------------------------------------
// Launch
// ---------------------------------------------------------------------------
extern "C" void kernel_launch(void* const* d_in, const int* in_sizes, int n_in,
                              void* d_out, int out_size, void* d_ws, size_t ws_size,
                              hipStream_t stream) {
    const float* x  = (const float*)d_in[0];
    const int*   ei = (const int*)  d_in[1];
    const float* W0 = (const float*)d_in[2];
    const float* b0 = (const float*)d_in[3];
    const float* W1 = (const float*)d_in[4];
    const float* b1 = (const float*)d_in[5];
    const float* W2 = (const float*)d_in[6];
    const float* b2 = (const float*)d_in[7];

    const int N = in_sizes[0] / HDIM;
    const int E = in_sizes[1] / 2;
    const int* src = ei;          // edge_index row 0
    const int* dst = ei + E;      // edge_index row 1

    float* ws   = (float*)d_ws;
    float* dinv = ws;                         // N floats (deg, then deg^{-1/2})
    float* norm = dinv + N;                   // E floats
    float* bufA = norm + E;                   // N*128 floats (h = x@W)
    float* bufB = bufA + (size_t)N * HDIM;    // N*128 floats (aggregation / activations)
    float* out  = (float*)d_out;

    const size_t NH = (size_t)N * HDIM;
    const dim3 blk(256);
    const int gN   = (N + 255) / 256;
    const int gE   = (E + 255) / 256;
    const int gNH  = (int)((NH + 255) / 256);
    const int gGem = (N + 127) / 128;
    const int gSc  = (E + 7) / 8;             // one wave per edge

    // Normalization: deg -> dinv -> per-edge norm
    zero_kernel<<<gN, blk, 0, stream>>>(dinv, (size_t)N);
    count_deg_kernel<<<gE, blk, 0, stream>>>(dst, dinv, E);
    dinv_kernel<<<gN, blk, 0, stream>>>(dinv, N);
    norm_kernel<<<gE, blk, 0, stream>>>(src, dst, dinv, norm, E);

    // Layer 0: x -> bufA (GEMM); aggregate bufA -> bufB; finalize bufB
    gcn_gemm_wmma<<<gGem, blk, 0, stream>>>(x, W0, bufA, N);
    zero_kernel<<<gNH, blk, 0, stream>>>(bufB, NH);
    scatter_add_kernel<<<gSc, blk, 0, stream>>>(bufA, src, dst, norm, bufB, E);
    finalize_kernel<<<gNH, blk, 0, stream>>>(bufB, bufA, dinv, b0, N);

    // Layer 1: bufB -> bufA (GEMM); aggregate -> bufB (safe: GEMM read it fully)
    gcn_gemm_wmma<<<gGem, blk, 0, stream>>>(bufB, W1, bufA, N);
    zero_kernel<<<gNH, blk, 0, stream>>>(bufB, NH);
    scatter_add_kernel<<<gSc, blk, 0, stream>>>(bufA, src, dst, norm, bufB, E);
    finalize_kernel<<<gNH, blk, 0, stream>>>(bufB, bufA, dinv, b1, N);

    // Layer 2: bufB -> bufA (GEMM); aggregate -> d_out; finalize d_out
    gcn_gemm_wmma<<<gGem, blk, 0, stream>>>(bufB, W2, bufA, N);
    zero_kernel<<<gNH, blk, 0, stream>>>(out, NH);
    scatter_add_kernel<<<gSc, blk, 0, stream>>>(bufA, src, dst, norm, out, E);
    finalize_kernel<<<gNH, blk, 0, stream>>>(out, bufA, dinv, b2, N);
}